// GCN_65609920414440
// MI455X (gfx1250) — compile-verified
//
#include <hip/hip_runtime.h>
#include <math.h>

typedef __attribute__((ext_vector_type(2))) float v2f;
typedef __attribute__((ext_vector_type(8))) float v8f;

#define TPB 256
#define EPSBN 1e-5f

// ---------------------------------------------------------------- utilities

__global__ void fill_kernel(float* __restrict__ p, float v, int n) {
    int i = blockIdx.x * blockDim.x + threadIdx.x;
    if (i < n) p[i] = v;
}

__global__ void deg_kernel(const int* __restrict__ dst, float* __restrict__ deg, int E) {
    int e = blockIdx.x * blockDim.x + threadIdx.x;
    if (e < E) atomicAdd(&deg[dst[e]], 1.0f);
}

__global__ void rsqrt_kernel(float* __restrict__ p, int n) {
    int i = blockIdx.x * blockDim.x + threadIdx.x;
    if (i < n) p[i] = rsqrtf(p[i]);
}

// ---------------------------------------------------------------- WMMA GEMM
// out[N, Ncols] = X[N,128] @ W[Ncols,128]^T  (W row-major [out,in], PyTorch style)
// mode: 0 = plain store, 1 = +bias & relu, 2 = +bias
// One wave computes a 16x16 tile with V_WMMA_F32_16X16X4_F32, K-loop of 32.
__global__ void wmma_gemm_kernel(const float* __restrict__ X,
                                 const float* __restrict__ W,
                                 const float* __restrict__ bias,
                                 float* __restrict__ out,
                                 int N, int Ncols, int mode) {
    const int K = 128;
    int wave = (int)((blockIdx.x * blockDim.x + threadIdx.x) >> 5);
    int lane = threadIdx.x & 31;
    int tilesPerRow = Ncols >> 4;
    int rowTile = wave / tilesPerRow;
    int colTile = wave - rowTile * tilesPerRow;
    if (rowTile * 16 >= N) return;   // wave-uniform: EXEC stays all-ones for WMMA

    int i0 = rowTile << 4;
    int j0 = colTile << 4;
    int m    = lane & 15;
    int half = lane >> 4;

    // A frag: lane l holds X[i0 + (l&15)][kk + 2*(l>>4) + {0,1}]
    // B frag: lane l holds W[j0 + (l&15)][kk + 2*(l>>4) + {0,1}]  (== B[k][n] of 4x16)
    const float* arow = X + (size_t)(i0 + m) * K + 2 * half;
    const float* brow = W + (size_t)(j0 + m) * K + 2 * half;

    v8f c = {0.f, 0.f, 0.f, 0.f, 0.f, 0.f, 0.f, 0.f};
#pragma unroll
    for (int kk = 0; kk < K; kk += 4) {
        v2f a = *(const v2f*)(arow + kk);
        v2f b = *(const v2f*)(brow + kk);
        c = __builtin_amdgcn_wmma_f32_16x16x4_f32(
                /*neg_a=*/false, a, /*neg_b=*/false, b,
                /*c_mod=*/(short)0, c, /*reuse_a=*/false, /*reuse_b=*/false);
    }

    // C/D layout: VGPR r, lanes 0-15 -> (M=r, N=lane); lanes 16-31 -> (M=r+8, N=lane-16)
    int orow = i0 + 8 * half;
    int ocol = j0 + m;
    float bv = (mode != 0) ? bias[ocol] : 0.0f;
#pragma unroll
    for (int r = 0; r < 8; ++r) {
        float v = c[r] + bv;
        if (mode == 1) v = fmaxf(v, 0.0f);
        out[(size_t)(orow + r) * Ncols + ocol] = v;
    }
}

// ---------------------------------------------------------------- GCN aggregation
// out[i][:] = bias[:] + hlin[i][:] * dinv[i]^2      (self-loop + bias folded in)
__global__ void agg_init_kernel(const float* __restrict__ hlin,
                                const float* __restrict__ dinv,
                                const float* __restrict__ bias,
                                float* __restrict__ out, int N) {
    int idx = blockIdx.x * blockDim.x + threadIdx.x;      // N*32 float4 slots
    if (idx >= N * 32) return;
    int i = idx >> 5, c4 = idx & 31;
    float di = dinv[i];
    float w  = di * di;
    float4 h4 = ((const float4*)(hlin + (size_t)i * 128))[c4];
    float4 b4 = ((const float4*)bias)[c4];
    float4 o;
    o.x = b4.x + h4.x * w;  o.y = b4.y + h4.y * w;
    o.z = b4.z + h4.z * w;  o.w = b4.w + h4.w * w;
    ((float4*)(out + (size_t)i * 128))[c4] = o;
}

// one wave per edge: coalesced float4 gather of h[src], 4 f32 atomics into out[dst]
__global__ void agg_edge_kernel(const float* __restrict__ hlin,
                                const float* __restrict__ dinv,
                                const int* __restrict__ src,
                                const int* __restrict__ dst,
                                float* __restrict__ out, int E) {
    int gid = blockIdx.x * blockDim.x + threadIdx.x;
    int e = gid >> 5;
    if (e >= E) return;
    int lane = threadIdx.x & 31;
    int s = src[e], d = dst[e];
    float norm = dinv[s] * dinv[d];
    float4 h4 = ((const float4*)(hlin + (size_t)s * 128))[lane];
    float* o = out + (size_t)d * 128 + lane * 4;
    atomicAdd(o + 0, h4.x * norm);
    atomicAdd(o + 1, h4.y * norm);
    atomicAdd(o + 2, h4.z * norm);
    atomicAdd(o + 3, h4.w * norm);
}

// ---------------------------------------------------------------- BatchNorm
// stats[0..127]=sum, [128..255]=sumsq (must be pre-zeroed)
__global__ void bn_stats_kernel(const float* __restrict__ h, float* __restrict__ stats,
                                int N, int rowsPerBlk) {
    __shared__ float sm[512];
    int tid = threadIdx.x;
    int c = tid & 127, half = tid >> 7;
    int r0 = blockIdx.x * rowsPerBlk;
    int r1 = r0 + rowsPerBlk; if (r1 > N) r1 = N;
    float s = 0.f, q = 0.f;
    for (int r = r0 + half; r < r1; r += 2) {      // coalesced: consecutive tid -> consecutive cols
        float v = h[(size_t)r * 128 + c];
        s += v; q += v * v;
    }
    sm[tid] = s; sm[256 + tid] = q;
    __syncthreads();
    if (tid < 128) {
        atomicAdd(&stats[c],       sm[tid]       + sm[tid + 128]);
        atomicAdd(&stats[128 + c], sm[256 + tid] + sm[384 + tid]);
    }
}

// stats[256..383]=scale, [384..511]=shift
__global__ void bn_finalize_kernel(float* __restrict__ stats,
                                   const float* __restrict__ g,
                                   const float* __restrict__ be, int N) {
    int c = threadIdx.x;        // 128 threads
    float invN = 1.0f / (float)N;
    float mu  = stats[c] * invN;
    float var = stats[128 + c] * invN - mu * mu;
    float sc  = g[c] * rsqrtf(var + EPSBN);
    stats[256 + c] = sc;
    stats[384 + c] = be[c] - mu * sc;
}

__global__ void bn_apply_kernel(float* __restrict__ h, const float* __restrict__ stats,
                                int N, int relu) {
    int idx = blockIdx.x * blockDim.x + threadIdx.x;
    if (idx >= N * 32) return;
    int i = idx >> 5, c4 = idx & 31;
    float4 v  = ((float4*)(h + (size_t)i * 128))[c4];
    float4 sc = ((const float4*)(stats + 256))[c4];
    float4 sh = ((const float4*)(stats + 384))[c4];
    v.x = v.x * sc.x + sh.x;  v.y = v.y * sc.y + sh.y;
    v.z = v.z * sc.z + sh.z;  v.w = v.w * sc.w + sh.w;
    if (relu) {
        v.x = fmaxf(v.x, 0.f); v.y = fmaxf(v.y, 0.f);
        v.z = fmaxf(v.z, 0.f); v.w = fmaxf(v.w, 0.f);
    }
    ((float4*)(h + (size_t)i * 128))[c4] = v;
}

// ---------------------------------------------------------------- log_softmax (64 cols)
__global__ void logsoftmax_kernel(const float* __restrict__ in, float* __restrict__ out, int N) {
    int gid = blockIdx.x * blockDim.x + threadIdx.x;
    int row = gid >> 5;
    if (row >= N) return;
    int lane = threadIdx.x & 31;
    const float* p = in + (size_t)row * 64;
    float a = p[lane], b = p[lane + 32];
    float m = fmaxf(a, b);
#pragma unroll
    for (int off = 16; off > 0; off >>= 1) m = fmaxf(m, __shfl_xor(m, off, 32));
    float s = __expf(a - m) + __expf(b - m);
#pragma unroll
    for (int off = 16; off > 0; off >>= 1) s += __shfl_xor(s, off, 32);
    float ls = __logf(s) + m;
    out[(size_t)row * 64 + lane]      = a - ls;
    out[(size_t)row * 64 + lane + 32] = b - ls;
}

// ---------------------------------------------------------------- launcher

extern "C" void kernel_launch(void* const* d_in, const int* in_sizes, int n_in,
                              void* d_out, int out_size, void* d_ws, size_t ws_size,
                              hipStream_t stream) {
    const float* x   = (const float*)d_in[0];
    const int*   ei  = (const int*)d_in[1];
    const float* W1  = (const float*)d_in[3];
    const float* b1  = (const float*)d_in[4];
    const float* g1  = (const float*)d_in[5];
    const float* be1 = (const float*)d_in[6];
    const float* W2  = (const float*)d_in[7];
    const float* b2  = (const float*)d_in[8];
    const float* g2  = (const float*)d_in[9];
    const float* be2 = (const float*)d_in[10];
    const float* Wh  = (const float*)d_in[11];
    const float* bh  = (const float*)d_in[12];
    const float* g3  = (const float*)d_in[13];
    const float* be3 = (const float*)d_in[14];
    const float* Wf  = (const float*)d_in[15];
    const float* bf  = (const float*)d_in[16];

    const int N = in_sizes[0] / 128;
    const int E = in_sizes[1] / 2;
    const int* src = ei;
    const int* dst = ei + E;

    float* ws    = (float*)d_ws;
    size_t nA    = (size_t)N * 128;
    float* bufA  = ws;                 // N x 128
    float* bufB  = ws + nA;            // N x 128
    float* dinv  = ws + 2 * nA;        // N
    float* stats = dinv + N;           // 512

    auto cdiv = [](long a, long b) { return (int)((a + b - 1) / b); };

    // --- symmetric degree normalization: deg = 1 + indeg; dinv = deg^-1/2
    fill_kernel<<<cdiv(N, TPB), TPB, 0, stream>>>(dinv, 1.0f, N);
    deg_kernel<<<cdiv(E, TPB), TPB, 0, stream>>>(dst, dinv, E);
    rsqrt_kernel<<<cdiv(N, TPB), TPB, 0, stream>>>(dinv, N);

    int rowsPerBlk = (N + 127) / 128;
    auto run_bn = [&](float* buf, const float* g, const float* be, int relu) {
        fill_kernel<<<1, TPB, 0, stream>>>(stats, 0.0f, 256);
        bn_stats_kernel<<<128, TPB, 0, stream>>>(buf, stats, N, rowsPerBlk);
        bn_finalize_kernel<<<1, 128, 0, stream>>>(stats, g, be, N);
        bn_apply_kernel<<<cdiv((long)N * 32, TPB), TPB, 0, stream>>>(buf, stats, N, relu);
    };

    auto run_gemm = [&](const float* X, const float* W, const float* bias,
                        float* out, int Ncols, int mode) {
        long waves = (long)((N + 15) / 16) * (Ncols >> 4);
        wmma_gemm_kernel<<<cdiv(waves * 32, TPB), TPB, 0, stream>>>(X, W, bias, out, N, Ncols, mode);
    };

    auto run_gcn = [&](const float* in, const float* W, const float* b, float* lin, float* outb) {
        run_gemm(in, W, nullptr, lin, 128, 0);                                   // lin = in @ W^T
        agg_init_kernel<<<cdiv((long)N * 32, TPB), TPB, 0, stream>>>(lin, dinv, b, outb, N);
        agg_edge_kernel<<<cdiv((long)E * 32, TPB), TPB, 0, stream>>>(lin, dinv, src, dst, outb, E);
    };

    // layer 1: GCN -> BN+ReLU       (result in bufB)
    run_gcn(x, W1, b1, bufA, bufB);
    run_bn(bufB, g1, be1, /*relu=*/1);

    // layer 2: GCN -> BN+ReLU       (result in bufB)
    run_gcn(bufB, W2, b2, bufA, bufB);
    run_bn(bufB, g2, be2, /*relu=*/1);

    // hidden dense: relu(bufB @ Wh^T + bh) -> BN (no relu)   (result in bufA)
    run_gemm(bufB, Wh, bh, bufA, 128, /*mode=*/1);
    run_bn(bufA, g3, be3, /*relu=*/0);

    // final dense: bufB[N,64] = bufA @ Wf^T + bf
    run_gemm(bufA, Wf, bf, bufB, 64, /*mode=*/2);

    // log_softmax rows of 64 -> d_out
    logsoftmax_kernel<<<cdiv((long)N * 32, TPB), TPB, 0, stream>>>(bufB, (float*)d_out, N);
}